// Cul_cor_13546326851762
// MI455X (gfx1250) — compile-verified
//
#include <hip/hip_runtime.h>
#include <math.h>

// Distance-correlation (Cul_cor, ind='distance') for x:[8,4096] f32.
// Two-pass fused formulation (never materializes the C x C matrices):
//   pass 1: m[f][i] = mean_j sqrt((x_fi - x_fj)^2 + eps)         (row means)
//   pass 2: A = a - m_i - m_j + tm fed directly into
//           V_WMMA_F32_16X16X4_F32 as both A and B (identical VGPR layout
//           for A and B=A^T per the CDNA5 ISA f32 16x16x4 layouts), giving
//           G[k,l] = sum_p A_k[p] A_l[p] with exact-f32 accumulate.
//   finalize: S = G/C^2, dcov = sqrt(max(S,0)+eps), sum of upper-tri ratios.

#define NF 8
#define CH 4096
#define EPS 1e-8f
// LDS strides chosen for zero bank conflicts:
//  X stage  (f32):    4100 floats  -> dword stride %64 == 4
//  XM stage (float2): 4100 float2  -> dword stride %64 == 8 (b128/lane OK)
#define XSTRIDE  4100
#define XMSTRIDE 4100

constexpr int GRID   = 256;
constexpr int BLOCK  = 256;
constexpr int NWAVES = GRID * (BLOCK / 32);   // wave32: 8 waves / block

typedef __attribute__((ext_vector_type(2))) float v2f;
typedef __attribute__((ext_vector_type(8))) float v8f;

extern __shared__ char dyn_smem[];

// ---------------- pass 1: per-factor row means of the distance matrix ------
__global__ __launch_bounds__(BLOCK) void rowmean_kernel(
    const float* __restrict__ X, float* __restrict__ M) {
  float* Xs = (float*)dyn_smem;                       // [8][XSTRIDE]
  for (int idx = threadIdx.x; idx < NF * CH; idx += BLOCK)
    Xs[(idx >> 12) * XSTRIDE + (idx & (CH - 1))] = X[idx];
  __syncthreads();

  const int lane = threadIdx.x & 31;
  const int wave = threadIdx.x >> 5;
  const int f = lane & 7;        // factor
  const int q = lane >> 3;       // j-phase 0..3
  const float* xr = Xs + f * XSTRIDE;
  const int gw = blockIdx.x * (BLOCK / 32) + wave;

  for (int i = gw; i < CH; i += NWAVES) {
    const float xi = xr[i];
    float rs = 0.f;
#pragma unroll 8
    for (int j = 0; j < CH; j += 4) {
      float t = xi - xr[j + q];
      rs += __builtin_amdgcn_sqrtf(t * t + EPS);
    }
    // sum the 4 phases that share a factor (lanes differing in bits 3,4)
    rs += __shfl_xor(rs, 8, 32);
    rs += __shfl_xor(rs, 16, 32);
    if (lane < 8) M[lane * CH + i] = rs * (1.0f / CH);
  }
}

// ---------------- total means per factor -----------------------------------
__global__ __launch_bounds__(256) void tm_kernel(
    const float* __restrict__ M, float* __restrict__ TM) {
  const int lane = threadIdx.x & 31;
  const int w = threadIdx.x >> 5;                     // factor 0..7
  float s = 0.f;
  for (int i = lane; i < CH; i += 32) s += M[w * CH + i];
#pragma unroll
  for (int m = 16; m >= 1; m >>= 1) s += __shfl_xor(s, m, 32);
  if (lane == 0) TM[w] = s * (1.0f / CH);
}

// ---------------- pass 2: centered Gram via WMMA ---------------------------
__global__ __launch_bounds__(BLOCK) void gram_kernel(
    const float* __restrict__ X, const float* __restrict__ M,
    const float* __restrict__ TM, float* __restrict__ Gpart) {
  float2* XM = (float2*)dyn_smem;                                  // [8][XMSTRIDE]
  float*  Gs = (float*)(dyn_smem + NF * XMSTRIDE * sizeof(float2)); // [8][64]
  for (int idx = threadIdx.x; idx < NF * CH; idx += BLOCK) {
    int f = idx >> 12, j = idx & (CH - 1);
    XM[f * XMSTRIDE + j] = float2{X[idx], M[idx]};
  }
  __syncthreads();

  const int lane = threadIdx.x & 31;
  const int wave = threadIdx.x >> 5;
  const int fl = lane & 15;        // WMMA M/N index
  const int fe = fl & 7;           // real factor (padding lanes duplicate -> broadcast)
  const int hl = lane >> 4;        // lane half selects K positions {0,1} vs {2,3}
  const float tmf = TM[fe];
  const float2* xmr = XM + fe * XMSTRIDE;
  const int gw = blockIdx.x * (BLOCK / 32) + wave;

  v8f acc[4] = {};                 // 4 independent accumulator chains
  for (int i = gw; i < CH; i += NWAVES) {
    const float2 xmi = xmr[i];
    const float xi = xmi.x;
    const float c0 = tmf - xmi.y;  // A = a - m_j + (tm - m_i)
    for (int j = 0; j < CH; j += 16) {
#pragma unroll
      for (int qq = 0; qq < 4; ++qq) {
        const int jj = j + 4 * qq + 2 * hl;   // even -> 16B aligned b128
        const float4 v = *reinterpret_cast<const float4*>(xmr + jj);
        float t0 = xi - v.x;
        float t1 = xi - v.z;
        float a0 = __builtin_amdgcn_sqrtf(t0 * t0 + EPS) - v.y + c0;
        float a1 = __builtin_amdgcn_sqrtf(t1 * t1 + EPS) - v.w + c0;
        v2f av = {a0, a1};
        // A and B layouts are identical for B = A^T: pass same regs twice.
        acc[qq] = __builtin_amdgcn_wmma_f32_16x16x4_f32(
            false, av, false, av, (short)0, acc[qq], false, false);
      }
    }
  }
  v8f a = acc[0] + acc[1] + acc[2] + acc[3];
  // D layout: VGPR r = row M=r (lanes 0-15), lane n = col N. Keep 8x8 block.
  if (lane < 8) {
#pragma unroll
    for (int r = 0; r < 8; ++r) Gs[wave * 64 + r * 8 + lane] = a[r];
  }
  __syncthreads();
  if (threadIdx.x < 64) {
    float s = 0.f;
#pragma unroll
    for (int w = 0; w < 8; ++w) s += Gs[w * 64 + threadIdx.x];
    Gpart[blockIdx.x * 64 + threadIdx.x] = s;   // deterministic: no atomics
  }
}

// ---------------- finalize: 8x8 math --------------------------------------
__global__ __launch_bounds__(64) void final_kernel(
    const float* __restrict__ Gpart, float* __restrict__ out) {
  __shared__ float Gs[64];
  const int t = threadIdx.x;
  float s = 0.f;
  for (int b = 0; b < GRID; ++b) s += Gpart[b * 64 + t];
  Gs[t] = s;
  __syncthreads();
  if (t == 0) {
    const float invC2 = 1.0f / ((float)CH * (float)CH);
    float dcov[8][8];
    for (int k = 0; k < 8; ++k)
      for (int l = 0; l < 8; ++l) {
        float S = Gs[k * 8 + l] * invC2;
        dcov[k][l] = sqrtf(fmaxf(S, 0.f) + EPS);
      }
    float cor = 0.f;
    for (int k = 0; k < 8; ++k)
      for (int l = k + 1; l < 8; ++l)
        cor += dcov[k][l] / sqrtf(dcov[k][k] * dcov[l][l] + EPS);
    out[0] = cor;
  }
}

extern "C" void kernel_launch(void* const* d_in, const int* in_sizes, int n_in,
                              void* d_out, int out_size, void* d_ws, size_t ws_size,
                              hipStream_t stream) {
  (void)in_sizes; (void)n_in; (void)out_size; (void)ws_size;
  const float* X = (const float*)d_in[0];
  float* out = (float*)d_out;
  float* ws = (float*)d_ws;
  float* Mm = ws;                       // [8][4096] row means
  float* TM = ws + NF * CH;             // [8] total means (+pad)
  float* Gp = ws + NF * CH + 64;        // [GRID][64] Gram partials

  const size_t lds1 = (size_t)NF * XSTRIDE * sizeof(float);                     // 131200 B
  const size_t lds2 = (size_t)NF * XMSTRIDE * sizeof(float2) + 64 * 8 * 4;      // 264448 B (<320KB/WG)

  rowmean_kernel<<<GRID, BLOCK, lds1, stream>>>(X, Mm);
  tm_kernel<<<1, 256, 0, stream>>>(Mm, TM);
  gram_kernel<<<GRID, BLOCK, lds2, stream>>>(X, Mm, TM, Gp);
  final_kernel<<<1, 64, 0, stream>>>(Gp, out);
}